// DTHyperNet_38955353374954
// MI455X (gfx1250) — compile-verified
//
#include <hip/hip_runtime.h>
#include <hip/hip_bf16.h>
#include <math.h>

// ---------------------------------------------------------------------------
// DTHyperNet fused forward for MI455X (gfx1250, wave32, WMMA + async-to-LDS).
//
// Shapes: B=524288, F=32, H=64, C=10, NN=3, NL=4.
// Host-constant analysis: NODE_IDX == all zeros, PSIGN = [[0,0],[0,0],[0,1],[0,1]]
//   => out[b,c] = s^2*(lc0+lc1) + s(1-s)*(lc2+lc3), s = sigmoid(<softmax(fi0), x-fs0>)
//   => only node-0 rows (0..31) of w_fi/w_fs are live.
// BatchNorm stats via Gram trick: sum(h)=W*(sum x)+B*b ; sum(h^2)=w^T (X^T X) w + ...
//   => stats pass is a 32x32xB GEMM (WMMA) + colsum, not a full h materialization.
// All GEMMs use V_WMMA_F32_16X16X4_F32 (fp32 in/out, matches reference precision).
// Global->LDS staging in the main pass uses GLOBAL_LOAD_ASYNC_TO_LDS_B128
// (ASYNCcnt) to avoid the VGPR round-trip on the bandwidth-critical path.
// ---------------------------------------------------------------------------

typedef __attribute__((ext_vector_type(2))) float v2f;
typedef __attribute__((ext_vector_type(8))) float v8f;

#define B_ROWS      524288
#define STATS_BLOCKS 128          // 4 waves each -> 512 slabs
#define STATS_SLABS  512
#define ROWS_PER_SLAB (B_ROWS / STATS_SLABS)   // 1024
#define SLAB_F       1056         // 1024 Gram + 32 colsum floats per slab

static __device__ __forceinline__ v8f wmma_f32(v2f a, v2f b, v8f c) {
    // D = A(16x4) x B(4x16) + C(16x16), fp32 everywhere.
    return __builtin_amdgcn_wmma_f32_16x16x4_f32(false, a, false, b,
                                                 (short)0, c, false, false);
}

// Async global->LDS copy, 16 bytes per op.  VDST carries the workgroup-relative
// LDS byte address (low 32 bits of the flat shared pointer, per the gfx1250
// aperture rule); VADDR is the 64-bit global address.
static __device__ __forceinline__ void async_copy_b128(unsigned lds_byte_off,
                                                       const void* gaddr) {
    asm volatile("global_load_async_to_lds_b128 %0, %1, off"
                 :: "v"(lds_byte_off), "v"((unsigned long long)(uintptr_t)gaddr)
                 : "memory");
}
static __device__ __forceinline__ void wait_async0() {
    asm volatile("s_wait_asynccnt 0" ::: "memory");
}

// ---------------------------------------------------------------------------
// Kernel 1: per-slab partial Gram (X^T X, 32x32) + column sums of x.
// Lane layout for 16x16x4: A[m,k]: m=lane%16, k=2*(lane/16)+v.  B[k,n]: n=lane%16,
// k=2*(lane/16)+v.  Since G = X^T X, the A fragment (cols as rows) and the B
// fragment (cols as cols) are the SAME per-lane loads of x -> each x element is
// loaded exactly once per 4-row chunk.
// ---------------------------------------------------------------------------
__global__ __launch_bounds__(128) void dt_stats(const float* __restrict__ x,
                                                float* __restrict__ gpart) {
    const int tid  = threadIdx.x;
    const int w    = tid >> 5;
    const int lane = tid & 31;
    const int half = lane >> 4;
    const int lm   = lane & 15;

    const int  slab = blockIdx.x * 4 + w;
    const long rb0  = (long)slab * ROWS_PER_SLAB;

    v8f g00 = {}, g01 = {}, g10 = {}, g11 = {};
    float c0 = 0.0f, c1 = 0.0f;

    const float* xp = x + rb0 * 32;
    for (int it = 0; it < ROWS_PER_SLAB / 4; ++it) {
        const float* xr = xp + (it * 4 + 2 * half) * 32;
        v2f a0, a1;
        a0.x = xr[lm];          a0.y = xr[32 + lm];        // cols 0..15
        a1.x = xr[16 + lm];     a1.y = xr[48 + lm];        // cols 16..31
        c0 += a0.x + a0.y;
        c1 += a1.x + a1.y;
        g00 = wmma_f32(a0, a0, g00);   // G[0:16, 0:16]
        g01 = wmma_f32(a0, a1, g01);   // G[0:16,16:32]
        g10 = wmma_f32(a1, a0, g10);   // G[16:32,0:16]
        g11 = wmma_f32(a1, a1, g11);   // G[16:32,16:32]
    }

    // C/D layout: VGPR r holds row (r + 8*half), lanes%16 = column.
    float* gp = gpart + (size_t)slab * SLAB_F;
#pragma unroll
    for (int r = 0; r < 8; ++r) {
        const int row = r + 8 * half;
        gp[row * 32 + lm]             = g00[r];
        gp[row * 32 + 16 + lm]        = g01[r];
        gp[(16 + row) * 32 + lm]      = g10[r];
        gp[(16 + row) * 32 + 16 + lm] = g11[r];
    }
    // colsum: col c held by lanes {c, c+16}; fold halves.
    const float t0 = c0 + __shfl_xor(c0, 16);
    const float t1 = c1 + __shfl_xor(c1, 16);
    if (half == 0) {
        gp[1024 + lm]      = t0;
        gp[1024 + 16 + lm] = t1;
    }
}

// ---------------------------------------------------------------------------
// Kernel 2: deterministic reduction of 512 slabs; fold BN into scale/shift:
//   h_norm = h*scale[j] + shift[j],  scale = g/sqrt(var+eps), shift = b - mu*scale
// ---------------------------------------------------------------------------
__global__ __launch_bounds__(256) void dt_stats_reduce(const float* __restrict__ gpart,
                                                       const float* __restrict__ w_in,
                                                       const float* __restrict__ b_in,
                                                       const float* __restrict__ bn_g,
                                                       const float* __restrict__ bn_b,
                                                       float* __restrict__ scaleshift) {
    __shared__ float Gs[1024];
    __shared__ float cs[32];
    const int tid = threadIdx.x;

    for (int idx = tid; idx < SLAB_F; idx += 256) {
        float s = 0.0f;
        for (int p = 0; p < STATS_SLABS; ++p) s += gpart[(size_t)p * SLAB_F + idx];
        if (idx < 1024) Gs[idx] = s; else cs[idx - 1024] = s;
    }
    __syncthreads();

    if (tid < 64) {
        float wr[32];
#pragma unroll
        for (int a = 0; a < 32; ++a) wr[a] = w_in[tid * 32 + a];
        const float bj = b_in[tid];

        float sdot = 0.0f;
#pragma unroll
        for (int a = 0; a < 32; ++a) sdot += wr[a] * cs[a];
        float q = 0.0f;
        for (int a = 0; a < 32; ++a) {
            float t = 0.0f;
#pragma unroll
            for (int b2 = 0; b2 < 32; ++b2) t += Gs[a * 32 + b2] * wr[b2];
            q += wr[a] * t;
        }
        const float invB = 1.0f / (float)B_ROWS;
        const float mu   = sdot * invB + bj;
        const float eh2  = q * invB + 2.0f * bj * (sdot * invB) + bj * bj;
        const float var  = eh2 - mu * mu;
        const float inv  = rsqrtf(var + 1e-5f);
        const float sc   = bn_g[tid] * inv;
        scaleshift[tid]      = sc;
        scaleshift[64 + tid] = bn_b[tid] - mu * sc;
    }
}

// ---------------------------------------------------------------------------
// Kernel 3: fused main pass. 128 threads = 4 waves; 64 rows per block.
// Per wave (16-row strip): GEMM1 (16x64 = 4 Ntiles x 8 ksteps WMMA) -> BN+GELU
// on accumulators -> restage strip to LDS in A layout -> GEMM2 (N=112 padded:
// fi0|fs0|lc, 7 Ntiles x 16 ksteps WMMA) -> softmax/sigmoid via 16-lane
// shuffles -> leaf-coefficient weighted lc sum -> out.
// ---------------------------------------------------------------------------
__global__ __launch_bounds__(128) void dt_main(const float* __restrict__ x,
                                               const float* __restrict__ w_in,
                                               const float* __restrict__ b_in,
                                               const float* __restrict__ w_fi,
                                               const float* __restrict__ b_fi,
                                               const float* __restrict__ w_fs,
                                               const float* __restrict__ b_fs,
                                               const float* __restrict__ w_lc,
                                               const float* __restrict__ b_lc,
                                               const float* __restrict__ scaleshift,
                                               float* __restrict__ out) {
    __shared__ float xs[64 * 32];     // x tile            (8 KB)
    __shared__ float w1[64 * 32];     // w_in [h][f]       (8 KB)
    __shared__ float w2[64 * 112];    // W2   [k=h][n]     (28 KB)
    __shared__ float bias2[112];
    __shared__ float bin[64];
    __shared__ float scs[64], shf[64];
    __shared__ float hst[4][16 * 64]; // per-wave h strip, reused as lc buffer (16 KB)

    const int tid  = threadIdx.x;
    const int w    = tid >> 5;
    const int lane = tid & 31;
    const int half = lane >> 4;
    const int lm   = lane & 15;
    const long rowBase = (long)blockIdx.x * 64;

    // ---- async staging of x tile and w_in (8 KB each, B128 per op) ------
    {
        const unsigned xs_lds = (unsigned)(uintptr_t)&xs[0];
        const unsigned w1_lds = (unsigned)(uintptr_t)&w1[0];
        const char* xsrc = (const char*)(x + rowBase * 32);
        const char* wsrc = (const char*)w_in;
#pragma unroll
        for (int i = 0; i < 4; ++i) {
            const int e = (tid + i * 128) * 16;          // byte offset, 512 x 16B
            async_copy_b128(xs_lds + e, xsrc + e);
            async_copy_b128(w1_lds + e, wsrc + e);
        }
    }

    // ---- W2 transpose staging: float4 source reads, LDS scatter ---------
    // W2 columns: [0,32)=w_fi node0 rows, [32,64)=w_fs node0 rows, [64,104)=w_lc, pad 0.
    for (int i = tid; i < 112 * 16; i += 128) {
        const int n  = i >> 4;                 // 0..111  (output column)
        const int k4 = (i & 15) * 4;           // 0..60   (k group of 4)
        float4 v = make_float4(0.0f, 0.0f, 0.0f, 0.0f);
        if      (n < 32)  v = *(const float4*)(w_fi + n * 64 + k4);
        else if (n < 64)  v = *(const float4*)(w_fs + (n - 32) * 64 + k4);
        else if (n < 104) v = *(const float4*)(w_lc + (n - 64) * 64 + k4);
        w2[(k4 + 0) * 112 + n] = v.x;
        w2[(k4 + 1) * 112 + n] = v.y;
        w2[(k4 + 2) * 112 + n] = v.z;
        w2[(k4 + 3) * 112 + n] = v.w;
    }
    if (tid < 112) {
        float v = 0.0f;
        if      (tid < 32)  v = b_fi[tid];
        else if (tid < 64)  v = b_fs[tid - 32];
        else if (tid < 104) v = b_lc[tid - 64];
        bias2[tid] = v;
    }
    if (tid < 64) { bin[tid] = b_in[tid]; scs[tid] = scaleshift[tid]; shf[tid] = scaleshift[64 + tid]; }

    wait_async0();          // retire this wave's async-to-LDS copies
    __syncthreads();        // make all staging visible block-wide

    // ---- GEMM1: h = x @ w_in^T + b_in -----------------------------------
    v8f acc1[4];
#pragma unroll
    for (int tn = 0; tn < 4; ++tn) {
        const float b = bin[16 * tn + lm];         // bias depends on column only
#pragma unroll
        for (int r = 0; r < 8; ++r) acc1[tn][r] = b;
    }
    const float* xrow = &xs[(w * 16 + lm) * 32];   // A row m = lane%16
#pragma unroll
    for (int kk = 0; kk < 8; ++kk) {
        const int k0 = 4 * kk + 2 * half;          // A/B lane k = 2*(lane/16)+v
        v2f a; a.x = xrow[k0]; a.y = xrow[k0 + 1];
#pragma unroll
        for (int tn = 0; tn < 4; ++tn) {
            const float* wr = &w1[(16 * tn + lm) * 32 + k0];  // B[k,n]=w_in[n][k]
            v2f b; b.x = wr[0]; b.y = wr[1];
            acc1[tn] = wmma_f32(a, b, acc1[tn]);
        }
    }

    // ---- BN + exact GELU on accumulators, restage strip in row-major ----
    float* hrow = hst[w];
#pragma unroll
    for (int tn = 0; tn < 4; ++tn) {
        const int col = 16 * tn + lm;
        const float sc = scs[col], sh = shf[col];
#pragma unroll
        for (int r = 0; r < 8; ++r) {
            float v = acc1[tn][r] * sc + sh;
            v = 0.5f * v * (1.0f + erff(v * 0.70710678118654752f));
            hrow[(r + 8 * half) * 64 + col] = v;   // D layout: row = r+8*half
        }
    }
    __syncthreads();

    // ---- GEMM2: [fi0 | fs0 | lc] = h @ W2 + bias2 -----------------------
    v8f acc2[7];
#pragma unroll
    for (int tn = 0; tn < 7; ++tn) {
        const float b = bias2[16 * tn + lm];
#pragma unroll
        for (int r = 0; r < 8; ++r) acc2[tn][r] = b;
    }
#pragma unroll
    for (int kk = 0; kk < 16; ++kk) {
        const int k0 = 4 * kk + 2 * half;
        v2f a; a.x = hrow[lm * 64 + k0]; a.y = hrow[lm * 64 + k0 + 1];
#pragma unroll
        for (int tn = 0; tn < 7; ++tn) {
            v2f b;
            b.x = w2[k0 * 112 + 16 * tn + lm];
            b.y = w2[(k0 + 1) * 112 + 16 * tn + lm];
            acc2[tn] = wmma_f32(a, b, acc2[tn]);
        }
    }
    __syncthreads();

    // ---- softmax / sigmoid / leaf mix: row m lives in one 16-lane half --
    float* lcb = hst[w];                           // reuse strip as lc buffer
#pragma unroll
    for (int r = 0; r < 8; ++r) {
        const int m = r + 8 * half;
        const float f0 = acc2[0][r], f1 = acc2[1][r];        // fi cols 0..31
        float mx = fmaxf(f0, f1);
#pragma unroll
        for (int mk = 1; mk < 16; mk <<= 1) mx = fmaxf(mx, __shfl_xor(mx, mk));
        const float e0 = expf(f0 - mx), e1 = expf(f1 - mx);
        float s = e0 + e1;
#pragma unroll
        for (int mk = 1; mk < 16; mk <<= 1) s += __shfl_xor(s, mk);
        const float p0 = e0 / s, p1 = e1 / s;

        const float* xr2 = &xs[(w * 16 + m) * 32];
        float d = p0 * (xr2[lm] - acc2[2][r]) + p1 * (xr2[16 + lm] - acc2[3][r]);
#pragma unroll
        for (int mk = 1; mk < 16; mk <<= 1) d += __shfl_xor(d, mk);
        const float s0 = 1.0f / (1.0f + expf(-d));
        const float ca = s0 * s0;                  // leaves 0,1
        const float cb = s0 * (1.0f - s0);         // leaves 2,3

#pragma unroll
        for (int tt = 0; tt < 3; ++tt) {           // lc cols 64..103 -> j=0..39
            const int j = 16 * tt + lm;
            if (j < 40) {
                const float co = (j < 20) ? ca : cb;
                lcb[m * 40 + j] = acc2[4 + tt][r] * co;
            }
        }
    }
    __syncthreads();

    // ---- out[m,c] = sum over 4 leaves of pre-scaled lc ------------------
    const long orow = rowBase + w * 16;
    for (int i = lane; i < 160; i += 32) {
        const int m = i / 10, c = i - m * 10;
        const float o = lcb[m * 40 + c] + lcb[m * 40 + 10 + c] +
                        lcb[m * 40 + 20 + c] + lcb[m * 40 + 30 + c];
        out[(orow + m) * 10 + c] = o;
    }
}

// ---------------------------------------------------------------------------
extern "C" void kernel_launch(void* const* d_in, const int* in_sizes, int n_in,
                              void* d_out, int out_size, void* d_ws, size_t ws_size,
                              hipStream_t stream) {
    const float* x    = (const float*)d_in[0];
    const float* w_in = (const float*)d_in[1];
    const float* b_in = (const float*)d_in[2];
    const float* bn_g = (const float*)d_in[3];
    const float* bn_b = (const float*)d_in[4];
    const float* w_fi = (const float*)d_in[5];
    const float* b_fi = (const float*)d_in[6];
    const float* w_fs = (const float*)d_in[7];
    const float* b_fs = (const float*)d_in[8];
    const float* w_lc = (const float*)d_in[9];
    const float* b_lc = (const float*)d_in[10];
    float* out = (float*)d_out;

    float* gpart      = (float*)d_ws;                          // 512*1056 floats
    float* scaleshift = gpart + (size_t)STATS_SLABS * SLAB_F;  // 128 floats

    dt_stats<<<STATS_BLOCKS, 128, 0, stream>>>(x, gpart);
    dt_stats_reduce<<<1, 256, 0, stream>>>(gpart, w_in, b_in, bn_g, bn_b, scaleshift);
    dt_main<<<B_ROWS / 64, 128, 0, stream>>>(x, w_in, b_in, w_fi, b_fi, w_fs, b_fs,
                                             w_lc, b_lc, scaleshift, out);
}